// PieceAvgPool_64304250355852
// MI455X (gfx1250) — compile-verified
//
#include <hip/hip_runtime.h>
#include <hip/hip_bf16.h>

// Piecewise average pooling (PCNN) for MI455X (gfx1250).
// out[b, p*I + i] = sum_{l: mask[b,l]==p+1} x[b,i,l] / max(count_p, eps-rule)
//
// Strategy: per-batch segmented sum expressed as V_WMMA_F32_16X16X4_F32
// (A = 16 I-rows x 4 L-cols of x from LDS, B = one-hot(l -> piece) built from
// mask). x tiles are streamed Global->LDS with the Tensor Data Mover
// (tensor_load_to_lds, TENSORcnt), double buffered, with TDM row padding
// (2 DWORDs per 16-DWORD row) for bank-conflict-free ds_load_b64 A fragments.

typedef __attribute__((ext_vector_type(2))) float v2f;
typedef __attribute__((ext_vector_type(8))) float v8f;
typedef __attribute__((ext_vector_type(4))) unsigned int v4u;
typedef __attribute__((ext_vector_type(8))) unsigned int v8u;

#define B_DIM   512
#define I_EMBED 256
#define L_DIM   512
#define P_NUM   3
#define CHUNK   16                    // L-elements staged per TDM op (64B/row)
#define ROWSTRIDE 18                  // 16 data floats + 2 pad floats (TDM pad)
#define NCHUNK  (L_DIM / CHUNK)       // 32
#define THREADS 256                   // 8 waves of 32

// Issue one TDM 2D tile load: 256 rows x CHUNK f32, row stride L_DIM, into LDS
// at lds_byte_addr with 2-DWORD padding after every 16 DWORDs stored.
// D# layout per CDNA5 ISA ch. 8 (group0 128b, group1 256b, groups2/3 unused=0).
__device__ __forceinline__ void tdm_load_chunk(const float* src, unsigned lds_byte_addr) {
    unsigned long long ga = (unsigned long long)(uintptr_t)src;
    v4u g0;
    g0.x = 1u;                                   // count=1 (valid user descriptor)
    g0.y = lds_byte_addr;                        // lds_addr (bytes)
    g0.z = (unsigned)ga;                         // global_addr[31:0]
    g0.w = (unsigned)(ga >> 32) | (2u << 30);    // global_addr[56:32] | type=2
    v8u g1;
    g1.s0 = (2u << 16)                           // data_size = 4 bytes
          | (1u << 20)                           // pad_enable
          | (3u << 22)                           // pad_interval: 16 DWORDs (64B)
          | (1u << 25);                          // pad_amount: 2 DWORDs (8B)
    g1.s1 = ((unsigned)L_DIM & 0xffffu) << 16;                               // tensor_dim0[15:0]
    g1.s2 = ((unsigned)L_DIM >> 16) | (((unsigned)I_EMBED & 0xffffu) << 16); // dim0 hi | tensor_dim1 lo
    g1.s3 = ((unsigned)I_EMBED >> 16) | ((unsigned)CHUNK << 16);             // dim1 hi | tile_dim0
    g1.s4 = (unsigned)I_EMBED;                   // tile_dim1 = 256 rows (tile_dim2 = 0)
    g1.s5 = (unsigned)L_DIM;                     // tensor_dim0_stride lo32 (elements)
    g1.s6 = 0u;                                  // stride0 hi | tensor_dim1_stride lo (unused, 2D)
    g1.s7 = 0u;
    v4u gz = {0u, 0u, 0u, 0u};                   // groups 2/3 unused (2D tensor)
    asm volatile("tensor_load_to_lds %0, %1, %2, %3"
                 :
                 : "s"(g0), "s"(g1), "s"(gz), "s"(gz)
                 : "memory");
}

__global__ __launch_bounds__(THREADS)
void PieceAvgPool_64304250355852_kernel(const float* __restrict__ x,
                                        const int* __restrict__ mask,
                                        float* __restrict__ out) {
    __shared__ float lds_x[2][I_EMBED * ROWSTRIDE];   // 2 x 18KB double buffer
    __shared__ int   lds_mask[L_DIM];                 // 2KB mask row
    __shared__ int   lds_cnt[4];                      // piece lengths

    const int b    = blockIdx.x;
    const int tid  = threadIdx.x;
    const int lane = tid & 31;
    const int wave = tid >> 5;
    const int lo   = lane & 15;    // A: I-row within tile / B: N (piece col) / D: N
    const int hi   = lane >> 4;    // K-half selector (K = {2*hi, 2*hi+1})

    if (tid < 4) lds_cnt[tid] = 0;
    __syncthreads();

    // Stage mask row + piece-length histogram (coalesced, once per block).
    for (int l = tid; l < L_DIM; l += THREADS) {
        int m = mask[b * L_DIM + l];
        lds_mask[l] = m;
        if (m > 0) atomicAdd(&lds_cnt[m - 1], 1);
    }

    const float* xb = x + (size_t)b * I_EMBED * L_DIM;

    // Prologue: wave 0 kicks chunk 0 into buffer 0 (TENSORcnt tracked).
    if (tid < 32) {
        tdm_load_chunk(xb, (unsigned)(uintptr_t)&lds_x[0][0]);
    }
    __syncthreads();   // mask/histogram staged (TDM completion waited in-loop)

    // Each wave owns two 16-row I tiles -> two independent accumulator chains.
    const int i0a = wave * 32;
    const int i0b = i0a + 16;
    v8f acc0 = {0.f, 0.f, 0.f, 0.f, 0.f, 0.f, 0.f, 0.f};
    v8f acc1 = acc0;

    for (int c = 0; c < NCHUNK; ++c) {
        const int cur = c & 1;
        if (tid < 32) {
            if (c + 1 < NCHUNK) {
                // Prefetch next chunk into the other buffer (safe: all waves
                // finished reading it at the end-of-previous-iteration barrier),
                // then wait for the *current* chunk (in-order TDM completion).
                tdm_load_chunk(xb + (size_t)(c + 1) * CHUNK,
                               (unsigned)(uintptr_t)&lds_x[cur ^ 1][0]);
                __builtin_amdgcn_s_wait_tensorcnt(1);
            } else {
                __builtin_amdgcn_s_wait_tensorcnt(0);
            }
        }
        __syncthreads();   // chunk c resident in lds_x[cur] for all waves

        const float* xs   = &lds_x[cur][0];
        const int lbase   = c * CHUNK;
        #pragma unroll
        for (int ks = 0; ks < CHUNK / 4; ++ks) {
            const int l0 = ks * 4;
            const int lg = lbase + l0 + 2 * hi;
            // B fragment: one-hot(l -> piece). Same-address LDS reads across
            // the 16 lanes of a half -> broadcast, no bank conflict.
            int m0 = lds_mask[lg];
            int m1 = lds_mask[lg + 1];
            v2f bf;
            bf.x = (m0 == lo + 1) ? 1.0f : 0.0f;
            bf.y = (m1 == lo + 1) ? 1.0f : 0.0f;
            // A fragments: ds_load_b64, rows strided by 18 words -> conflict-free.
            v2f a0 = *(const v2f*)(xs + (i0a + lo) * ROWSTRIDE + l0 + 2 * hi);
            v2f a1 = *(const v2f*)(xs + (i0b + lo) * ROWSTRIDE + l0 + 2 * hi);
            // D(MxN) = A(16x4 of I x L) * B(4x16 of L x piece) + C, full f32.
            acc0 = __builtin_amdgcn_wmma_f32_16x16x4_f32(
                       false, a0, false, bf, (short)0, acc0, false, false);
            acc1 = __builtin_amdgcn_wmma_f32_16x16x4_f32(
                       false, a1, false, bf, (short)0, acc1, false, false);
        }
        __syncthreads();   // all waves done with lds_x[cur] before TDM reuses it
    }

    // Epilogue: D layout -> lane lo holds piece column lo (lo<3), vgpr r holds
    // I-row (i0 + r + 8*hi). Divide by piece length (0 -> 1e-5, sum is 0 anyway).
    if (lo < P_NUM) {
        int cnt     = lds_cnt[lo];
        float denom = (cnt == 0) ? 1e-5f : (float)cnt;
        float inv   = 1.0f / denom;
        float* ob = out + (size_t)b * (P_NUM * I_EMBED) + lo * I_EMBED;
        #pragma unroll
        for (int r = 0; r < 8; ++r) {
            ob[i0a + r + 8 * hi] = acc0[r] * inv;
            ob[i0b + r + 8 * hi] = acc1[r] * inv;
        }
    }
}

extern "C" void kernel_launch(void* const* d_in, const int* in_sizes, int n_in,
                              void* d_out, int out_size, void* d_ws, size_t ws_size,
                              hipStream_t stream) {
    (void)in_sizes; (void)n_in; (void)out_size; (void)d_ws; (void)ws_size;
    const float* x    = (const float*)d_in[0];
    const int*   mask = (const int*)d_in[1];
    float*       out  = (float*)d_out;
    dim3 grid(B_DIM);
    dim3 block(THREADS);
    PieceAvgPool_64304250355852_kernel<<<grid, block, 0, stream>>>(x, mask, out);
}